// DEER_MLP_10668698763979
// MI455X (gfx1250) — compile-verified
//
#include <hip/hip_runtime.h>

typedef __attribute__((ext_vector_type(2))) float v2f;
typedef __attribute__((ext_vector_type(8))) float v8f;
typedef __attribute__((ext_vector_type(4))) unsigned uint32x4;
typedef __attribute__((ext_vector_type(8))) int     int32x8;
typedef __attribute__((ext_vector_type(4))) int     int32x4;

#define LIF_TAU   2.0f
#define LIF_VTH   1.0f
#define LIF_DECAY 0.5f     // 1 - 1/tau
#define BN_EPS    1e-5f

// ---------------------------------------------------------------------------
// Tile geometry: block tile 128x64, 256 threads = 8 wave32 (4 M x 2 N),
// each wave computes 32x32 via 2x2 V_WMMA_F32_16X16X4_F32 fragments.
// K chunk = 16 (4 WMMA k-steps). TDM pads every 16 DWORDs with 1 DWORD,
// giving an LDS row stride of 17 floats (bank-conflict-free column reads).
// ---------------------------------------------------------------------------
#define BM   128
#define BN_  64
#define KC   16
#define LDT  17                 // LDS row stride in floats (16 data + 1 TDM pad)
#define A_TILE (BM * LDT)       // 2176 floats
#define B_TILE (BN_ * LDT)      // 1088 floats
#define BUF_SZ (A_TILE + B_TILE)

typedef __attribute__((address_space(3))) const void lds_cv;

__device__ __forceinline__ unsigned lds_addr_of(const void* p)
{
    // generic -> LDS addrspace cast; integer value of an AS(3) pointer is the
    // byte offset inside the wave's LDS allocation (what D#.lds_addr wants).
    return (unsigned)(__UINTPTR_TYPE__)(lds_cv*)p;
}

// One Tensor-Data-Mover load: 2-D tile (tileRows x 16 fp32), row stride
// rowStrideElems, global tile start -> LDS at ldsByte with 17-float row pitch.
__device__ __forceinline__ void tdm_load_tile(unsigned ldsByte, const float* tile,
                                              int tileRows, int rowStrideElems)
{
    const unsigned long long ga = (unsigned long long)(__UINTPTR_TYPE__)tile;

    uint32x4 g0;
    g0[0] = 1u;                                                // count=1 (valid), user mode
    g0[1] = ldsByte;                                           // lds_addr
    g0[2] = (unsigned)ga;                                      // global_addr[31:0]
    g0[3] = ((unsigned)(ga >> 32) & 0x01FFFFFFu) | 0x80000000u;// global_addr[56:32] | type=2

    int32x8 g1;
    g1[0] = 0x00D20000;            // data_size=4B(2<<16), pad_enable(1<<20), pad_interval=3(16DW), pad_amount=0(1DW)
    g1[1] = 0;                     // atomic_barrier_addr=0, tensor_dim0[15:0] (dim0 = 1<<30)
    g1[2] = 0x4000;                // tensor_dim0[31:16]=0x4000, tensor_dim1[15:0]=0 (dim1 = 1<<30)
    g1[3] = (16 << 16) | 0x4000;   // tile_dim0=16 | tensor_dim1[31:16]
    g1[4] = tileRows;              // tile_dim1 (tile_dim2=0 -> 2-D tile)
    g1[5] = rowStrideElems;        // tensor_dim0_stride[31:0]
    g1[6] = 0;                     // stride[47:32]=0, tensor_dim1_stride[15:0]=0
    g1[7] = 0;

    const int32x4 z4 = {0, 0, 0, 0};
#if __clang_major__ >= 23
    const int32x8 z8 = {0, 0, 0, 0, 0, 0, 0, 0};
    __builtin_amdgcn_tensor_load_to_lds(g0, g1, z4, z4, z8, 0);
#else
    __builtin_amdgcn_tensor_load_to_lds(g0, g1, z4, z4, 0);
#endif
}

// ---------------------------------------------------------------------------
// WMMA fp32 GEMM:  Y[M x N] = A[M x K] * B[N x K]^T + bias[N]   (row-major)
// TDM double-buffered tiles; wave 0 issues DMA for chunk i+1 before the
// compute of chunk i; one s_wait_tensorcnt + one barrier per chunk.
// ---------------------------------------------------------------------------
__global__ __launch_bounds__(256)
void gemm_wmma_f32(const float* __restrict__ A, const float* __restrict__ B,
                   const float* __restrict__ bias, float* __restrict__ Y,
                   int M, int N, int K)
{
    __shared__ float smem[BUF_SZ * 2];

    const int tid   = threadIdx.x;
    const int lane  = tid & 31;          // wave32
    const int wave  = tid >> 5;          // 0..7
    const int waveN = wave & 1;          // 0..1
    const int waveM = wave >> 1;         // 0..3

    const int blockM = blockIdx.y * BM;
    const int blockN = blockIdx.x * BN_;

    const int l16   = lane & 15;
    const int lHalf = lane >> 4;         // 0 | 1
    const int kl    = lHalf * 2;         // this half-wave's K pair offset

    const unsigned ldsBase = lds_addr_of(smem);

    v8f c00 = {}, c01 = {}, c10 = {}, c11 = {};

    const float* Abase = A + (size_t)blockM * K;
    const float* Bbase = B + (size_t)blockN * K;
    const int nk = K / KC;

    // prologue: DMA chunk 0 into buffer 0
    if (wave == 0) {
        tdm_load_tile(ldsBase,              Abase, BM,  K);
        tdm_load_tile(ldsBase + A_TILE * 4, Bbase, BN_, K);
    }
    __builtin_amdgcn_s_wait_tensorcnt(0);
    __syncthreads();

    for (int i = 0; i < nk; ++i) {
        const int cur = i & 1;

        // prefetch next chunk into the other buffer (overlaps with WMMA below)
        if (wave == 0 && (i + 1) < nk) {
            const unsigned dst = ldsBase + (unsigned)((1 - cur) * BUF_SZ * 4);
            tdm_load_tile(dst,              Abase + (size_t)(i + 1) * KC, BM,  K);
            tdm_load_tile(dst + A_TILE * 4, Bbase + (size_t)(i + 1) * KC, BN_, K);
        }

        const float* As_ = smem + cur * BUF_SZ;
        const float* Bs_ = As_ + A_TILE;

        #pragma unroll
        for (int k4 = 0; k4 < KC; k4 += 4) {
            const int ka = k4 + kl;
            v2f a0, a1, b0, b1;
            const float* arow0 = &As_[(waveM * 32 + l16) * LDT + ka];
            const float* arow1 = &As_[(waveM * 32 + 16 + l16) * LDT + ka];
            a0.x = arow0[0]; a0.y = arow0[1];
            a1.x = arow1[0]; a1.y = arow1[1];
            const float* brow0 = &Bs_[(waveN * 32 + l16) * LDT + ka];
            const float* brow1 = &Bs_[(waveN * 32 + 16 + l16) * LDT + ka];
            b0.x = brow0[0]; b0.y = brow0[1];
            b1.x = brow1[0]; b1.y = brow1[1];

            c00 = __builtin_amdgcn_wmma_f32_16x16x4_f32(false, a0, false, b0, (short)0, c00, false, false);
            c01 = __builtin_amdgcn_wmma_f32_16x16x4_f32(false, a0, false, b1, (short)0, c01, false, false);
            c10 = __builtin_amdgcn_wmma_f32_16x16x4_f32(false, a1, false, b0, (short)0, c10, false, false);
            c11 = __builtin_amdgcn_wmma_f32_16x16x4_f32(false, a1, false, b1, (short)0, c11, false, false);
        }

        // next buffer ready (wave0's DMA done; no-op wait for other waves),
        // and all waves done reading the current buffer.
        __builtin_amdgcn_s_wait_tensorcnt(0);
        __syncthreads();
    }

    // Epilogue: C/D layout -> reg r holds (M = r + 8*lHalf, N = lane&15)
    #pragma unroll
    for (int r = 0; r < 8; ++r) {
        const int mloc = r + 8 * lHalf;
        const int gm0  = blockM + waveM * 32 + mloc;
        const int gm1  = gm0 + 16;
        const int gn0  = blockN + waveN * 32 + l16;
        const int gn1  = gn0 + 16;
        const float bi0 = bias[gn0];
        const float bi1 = bias[gn1];
        Y[(size_t)gm0 * N + gn0] = c00[r] + bi0;
        Y[(size_t)gm0 * N + gn1] = c01[r] + bi1;
        Y[(size_t)gm1 * N + gn0] = c10[r] + bi0;
        Y[(size_t)gm1 * N + gn1] = c11[r] + bi1;
    }
}

// ---------------------------------------------------------------------------
// Per-channel BatchNorm statistics: stats[ch] += sum, stats[chans+ch] += sum^2
// ---------------------------------------------------------------------------
__global__ __launch_bounds__(256)
void bn_stats(const float* __restrict__ Y, float* __restrict__ stats,
              int rows, int chans)
{
    const int ch = blockIdx.x * 256 + threadIdx.x;
    float s = 0.f, s2 = 0.f;
    for (int r = blockIdx.y; r < rows; r += gridDim.y) {
        const float v = Y[(size_t)r * chans + ch];
        s  += v;
        s2 += v * v;
    }
    atomicAdd(&stats[ch], s);
    atomicAdd(&stats[chans + ch], s2);
}

// ---------------------------------------------------------------------------
// Fused BatchNorm (training-mode stats) + sequential 4-step LIF.
// DEER Newton with MAX_ITER=10 >= T=4 converges exactly to this recurrence.
// ---------------------------------------------------------------------------
__global__ __launch_bounds__(256)
void bn_lif(const float* __restrict__ Yin, float* __restrict__ Sout,
            const float* __restrict__ stats,
            const float* __restrict__ gamma, const float* __restrict__ beta,
            int Mb, int chans)
{
    const int idx = blockIdx.x * 256 + threadIdx.x;   // m*chans + ch
    const int ch  = idx % chans;

    const float inv   = 1.0f / (float)(4 * Mb);
    const float mean  = stats[ch] * inv;
    const float var   = stats[chans + ch] * inv - mean * mean;
    const float scale = rsqrtf(var + BN_EPS) * gamma[ch];
    const float shift = beta[ch] - mean * scale;

    const size_t stride = (size_t)Mb * (size_t)chans;
    float v = 0.0f;
    #pragma unroll
    for (int t = 0; t < 4; ++t) {
        const float y  = Yin[(size_t)idx + (size_t)t * stride];
        const float yn = y * scale + shift;                 // BatchNorm
        const float h  = LIF_DECAY * v + yn * (1.0f / LIF_TAU);
        const float sp = (h >= LIF_VTH) ? 1.0f : 0.0f;      // Heaviside fwd
        v = h * (1.0f - sp);                                // hard reset
        Sout[(size_t)idx + (size_t)t * stride] = sp;
    }
}

__global__ __launch_bounds__(256)
void zero_f32(float* __restrict__ p, int n)
{
    const int i = blockIdx.x * 256 + threadIdx.x;
    if (i < n) p[i] = 0.0f;
}

// ---------------------------------------------------------------------------
extern "C" void kernel_launch(void* const* d_in, const int* in_sizes, int n_in,
                              void* d_out, int out_size, void* d_ws, size_t ws_size,
                              hipStream_t stream)
{
    (void)in_sizes; (void)n_in; (void)out_size; (void)ws_size;

    const float* x   = (const float*)d_in[0];  // (T,B,N,C) = (4,32,196,384)
    const float* W1  = (const float*)d_in[1];  // (H,C) = (1536,384)
    const float* b1  = (const float*)d_in[2];
    const float* g1  = (const float*)d_in[3];
    const float* be1 = (const float*)d_in[4];
    const float* W2  = (const float*)d_in[5];  // (C,H) = (384,1536)
    const float* b2  = (const float*)d_in[6];
    const float* g2  = (const float*)d_in[7];
    const float* be2 = (const float*)d_in[8];
    float* out = (float*)d_out;

    const int C  = 384, H = 1536;
    const int R  = 4 * 32 * 196;   // 25088 rows (= T*B*N)
    const int Mb = 32 * 196;       // 6272 (= B*N, LIF batch)

    // workspace layout (fp32): Y1[R*H] | Y2[R*C] | stats1[2H] | stats2[2C]
    float* Y1  = (float*)d_ws;
    float* Y2  = Y1 + (size_t)R * H;
    float* st1 = Y2 + (size_t)R * C;
    float* st2 = st1 + 2 * H;
    (void)st2;

    // 1) zero BN statistics (st1 and st2 contiguous: 2H + 2C floats)
    {
        const int n = 2 * (H + C);
        zero_f32<<<(n + 255) / 256, 256, 0, stream>>>(st1, n);
    }
    // 2) FC1: Y1 = x * W1^T + b1          (25088 x 1536, K=384)
    gemm_wmma_f32<<<dim3(H / BN_, R / BM), 256, 0, stream>>>(x, W1, b1, Y1, R, H, C);
    // 3) BN1 stats
    bn_stats<<<dim3(H / 256, 64), 256, 0, stream>>>(Y1, st1, R, H);
    // 4) BN1 + LIF1 (in-place spikes)
    bn_lif<<<(Mb * H) / 256, 256, 0, stream>>>(Y1, Y1, st1, g1, be1, Mb, H);
    // 5) FC2: Y2 = s1 * W2^T + b2         (25088 x 384, K=1536)
    gemm_wmma_f32<<<dim3(C / BN_, R / BM), 256, 0, stream>>>(Y1, W2, b2, Y2, R, C, H);
    // 6) BN2 stats
    bn_stats<<<dim3(C / 256, 64), 256, 0, stream>>>(Y2, st2, R, C);
    // 7) BN2 + LIF2 -> output spikes
    bn_lif<<<(Mb * C) / 256, 256, 0, stream>>>(Y2, out, st2, g2, be2, Mb, C);
}